// EmissionCrfNerModel_81037442941490
// MI455X (gfx1250) — compile-verified
//
#include <hip/hip_runtime.h>
#include <stdint.h>

#define T_LEN 16384
#define L_TAGS 64
#define F_PER 14
#define BOS_TAG 0
#define EOS_TAG 1

// ---- TDM descriptor plumbing (CDNA5 Tensor Data Mover) -----------------
typedef uint32_t v4u __attribute__((ext_vector_type(4)));
typedef int      v8i __attribute__((ext_vector_type(8)));
typedef int      v4i __attribute__((ext_vector_type(4)));

// Build a 2-D D# (CDNA5 ISA) and issue TENSOR_LOAD_TO_LDS.
// dsz_log2: 0=1B, 1=2B, 2=4B elements. tile = tile0 x tile1, row stride
// stride0 (in elements). ga = byte address of tile start.
// This toolchain's builtin is the 6-arg form:
//   (v4u g0, v8i g1, v4i g2, v4i g3, v8i g4, i32 cpol)
__device__ __forceinline__ void tdm_load_2d(uint32_t lds_off, uint64_t ga,
                                            uint32_t dsz_log2,
                                            uint32_t tile0, uint32_t tile1,
                                            uint32_t td0, uint32_t td1,
                                            uint32_t stride0)
{
    v4u g0;
    g0[0] = 1u;                                   // count=1, is_restore=0
    g0[1] = lds_off;                              // lds_addr (bytes)
    g0[2] = (uint32_t)ga;                         // global_addr[31:0]
    g0[3] = ((uint32_t)(ga >> 32) & 0x1FFFFFFu)   // global_addr[56:32]
            | (2u << 30);                         // type=2 ("image")
    v8i g1;
    g1[0] = (int)(dsz_log2 << 16);                // wg_mask=0 | data_size
    g1[1] = (int)((td0 & 0xFFFFu) << 16);         // bar_addr=0 | tensor_dim0 lo
    g1[2] = (int)((td0 >> 16) | ((td1 & 0xFFFFu) << 16)); // dim0 hi | dim1 lo
    g1[3] = (int)((td1 >> 16) | (tile0 << 16));   // dim1 hi | tile_dim0
    g1[4] = (int)(tile1 & 0xFFFFu);               // tile_dim1 | tile_dim2=0
    g1[5] = (int)stride0;                         // tensor_dim0_stride[31:0]
    g1[6] = 0;                                    // stride0 hi | stride1 lo
    g1[7] = 0;                                    // stride1 hi
    v4i g2 = {0, 0, 0, 0};                        // 2-D: dims 2..4 unused
    v4i g3 = {0, 0, 0, 0};
    v8i g4 = {0, 0, 0, 0, 0, 0, 0, 0};            // unused (zero-filled)
    __builtin_amdgcn_tensor_load_to_lds(g0, g1, g2, g3, g4, 0);
}

__device__ __forceinline__ void wait_tensor0() {
    asm volatile("" ::: "memory");
    __builtin_amdgcn_s_wait_tensorcnt(0);
    asm volatile("" ::: "memory");
}

__device__ __forceinline__ float bcast_lane(float v, int lane) {
    return __int_as_float(__builtin_amdgcn_readlane(__float_as_int(v), lane));
}

// ---- K1: emissions = sum of 14 gathered weights per (t, tag) ------------
__global__ void emis_kernel(const int* __restrict__ fi,
                            const float* __restrict__ w,
                            float* __restrict__ em)
{
    int id = blockIdx.x * blockDim.x + threadIdx.x;      // id = t*64 + tag
    if (id >= T_LEN * L_TAGS) return;
    const int* p = fi + (long long)id * F_PER;
    float s = 0.f;
#pragma unroll
    for (int f = 0; f < F_PER; ++f) s += w[p[f]];
    em[id] = s;
}

// ---- K2: sequential Viterbi forward, single wave32 ----------------------
// Lane l owns destination states l and l+32. TDM double-buffers 64-row
// emission tiles into LDS while the VALU recurrence runs.
#define FCH 64
#define FNC (T_LEN / FCH)

__global__ void __launch_bounds__(32)
viterbi_fwd(const float* __restrict__ trans, const float* __restrict__ em,
            unsigned char* __restrict__ bp, float* __restrict__ out,
            int* __restrict__ ftag)
{
    __shared__ float s_em[2][FCH * L_TAGS];     // 32 KB double buffer
    __shared__ float s_red[L_TAGS];
    const int l = threadIdx.x;

    // transition columns pinned in VGPRs (128 f32 per lane)
    float tcol_lo[64], tcol_hi[64];
#pragma unroll
    for (int i = 0; i < 64; ++i) {
        tcol_lo[i] = trans[i * 64 + l];
        tcol_hi[i] = trans[i * 64 + l + 32];
    }

    uint32_t lds0 = (uint32_t)(uintptr_t)&s_em[0][0];
    uint32_t lds1 = (uint32_t)(uintptr_t)&s_em[1][0];
    uint64_t base = (uint64_t)(uintptr_t)em;

    tdm_load_2d(lds0, base, 2, L_TAGS, FCH, L_TAGS, T_LEN, L_TAGS);
    wait_tensor0();

    // alpha0 = trans[BOS][:] + em[0][:]
    float alpha_lo = trans[BOS_TAG * 64 + l]      + s_em[0][l];
    float alpha_hi = trans[BOS_TAG * 64 + l + 32] + s_em[0][l + 32];

    for (int c = 0; c < FNC; ++c) {
        if (c + 1 < FNC) {
            tdm_load_2d(((c + 1) & 1) ? lds1 : lds0,
                        base + (uint64_t)(c + 1) * FCH * L_TAGS * 4,
                        2, L_TAGS, FCH, L_TAGS, T_LEN, L_TAGS);
        }
        const float* buf = s_em[c & 1];
        for (int tl = (c == 0) ? 1 : 0; tl < FCH; ++tl) {
            int t = c * FCH + tl;
            float best_lo = -3.4e38f, best_hi = -3.4e38f;
            int bi_lo = 0, bi_hi = 0;
#pragma unroll
            for (int i = 0; i < 64; ++i) {
                float a = (i < 32) ? bcast_lane(alpha_lo, i)
                                   : bcast_lane(alpha_hi, i - 32);
                float s0 = a + tcol_lo[i];
                if (s0 > best_lo) { best_lo = s0; bi_lo = i; }
                float s1 = a + tcol_hi[i];
                if (s1 > best_hi) { best_hi = s1; bi_hi = i; }
            }
            alpha_lo = best_lo + buf[tl * 64 + l];
            alpha_hi = best_hi + buf[tl * 64 + l + 32];
            bp[(long long)t * 64 + l]      = (unsigned char)bi_lo;
            bp[(long long)t * 64 + l + 32] = (unsigned char)bi_hi;
        }
        wait_tensor0();
    }

    // end_scores = alpha + trans[:, EOS]; max/argmax over 64 states
    s_red[l]      = alpha_lo + trans[l * 64 + EOS_TAG];
    s_red[l + 32] = alpha_hi + trans[(l + 32) * 64 + EOS_TAG];
    __syncthreads();                 // single wave -> hardware NOP
    if (l == 0) {
        float best = s_red[0]; int bi = 0;
        for (int j = 1; j < 64; ++j)
            if (s_red[j] > best) { best = s_red[j]; bi = j; }
        out[0] = best;                          // max_final_scores
        out[1 + T_LEN - 1] = (float)bi;         // path[T-1] = final tag
        *ftag = bi;
    }
}

// ---- K3: backtrace, bp rows staged into LDS via TDM (reverse chunks) ----
#define BCH 256
#define BNC (T_LEN / BCH)

__global__ void __launch_bounds__(32)
viterbi_back(const unsigned char* __restrict__ bp,
             const int* __restrict__ ftag, float* __restrict__ out)
{
    __shared__ unsigned char s_bp[2][BCH * L_TAGS];   // 32 KB double buffer
    uint32_t lds0 = (uint32_t)(uintptr_t)&s_bp[0][0];
    uint32_t lds1 = (uint32_t)(uintptr_t)&s_bp[1][0];
    uint64_t base = (uint64_t)(uintptr_t)bp;

    tdm_load_2d(((BNC - 1) & 1) ? lds1 : lds0,
                base + (uint64_t)(BNC - 1) * BCH * L_TAGS,
                0, L_TAGS, BCH, L_TAGS, T_LEN, L_TAGS);
    wait_tensor0();

    int tag = *ftag;                 // only lane 0's copy matters
    for (int c = BNC - 1; c >= 0; --c) {
        if (c > 0) {
            tdm_load_2d(((c - 1) & 1) ? lds1 : lds0,
                        base + (uint64_t)(c - 1) * BCH * L_TAGS,
                        0, L_TAGS, BCH, L_TAGS, T_LEN, L_TAGS);
        }
        if (threadIdx.x == 0) {
            const unsigned char* buf = s_bp[c & 1];
            int tl_lo = (c == 0) ? 1 : 0;              // t >= 1
            for (int tl = BCH - 1; tl >= tl_lo; --tl) {
                int t = c * BCH + tl;
                tag = buf[tl * 64 + tag];              // LDS-latency chase
                out[t] = (float)tag;                   // d_out[1+(t-1)]
            }
        }
        wait_tensor0();
    }
}

// ---- launch -------------------------------------------------------------
extern "C" void kernel_launch(void* const* d_in, const int* in_sizes, int n_in,
                              void* d_out, int out_size, void* d_ws, size_t ws_size,
                              hipStream_t stream)
{
    const int*   fi    = (const int*)d_in[0];    // (T, L, 14) int32
    const float* w     = (const float*)d_in[1];  // (5M, 1)  f32
    const float* trans = (const float*)d_in[2];  // (64, 64) f32
    float* out = (float*)d_out;                  // [score, path(16384)]

    char* ws = (char*)d_ws;
    float*         em   = (float*)ws;                                   // 4 MB
    unsigned char* bpws = (unsigned char*)(ws + (size_t)T_LEN * 64 * 4); // 1 MB
    int*           ftag = (int*)(ws + (size_t)T_LEN * 64 * 4 + (size_t)T_LEN * 64);

    int n = T_LEN * L_TAGS;
    emis_kernel<<<(n + 255) / 256, 256, 0, stream>>>(fi, w, em);
    viterbi_fwd<<<1, 32, 0, stream>>>(trans, em, bpws, out, ftag);
    viterbi_back<<<1, 32, 0, stream>>>(bpws, ftag, out);
}